// DotProductAttention_87024627352347
// MI455X (gfx1250) — compile-verified
//
#include <hip/hip_runtime.h>

typedef float v2f __attribute__((ext_vector_type(2)));
typedef float v8f __attribute__((ext_vector_type(8)));

#define NEGVAL (-1000000.0f)

constexpr int D_DIM   = 128;        // head dim (fixed by problem)
constexpr int KC      = 32;         // keys per LDS chunk
constexpr int WAVES   = 8;          // waves per block (wave32)
constexpr int MT      = 16 * WAVES; // 128 query rows per block
constexpr int KSTRIDE = D_DIM + 4;  // padded LDS row stride (floats) -> conflict-free B reads
constexpr int PSTRIDE = KC + 4;     // padded P row stride (floats)

// ---- async global -> LDS copy of 16 bytes (CDNA5 ASYNCcnt path, no VGPR staging) ----
#if __has_builtin(__builtin_amdgcn_global_load_async_to_lds_b128)
typedef int i4v __attribute__((vector_size(16)));
typedef __attribute__((address_space(1))) i4v g_i4v;   // global int4
typedef __attribute__((address_space(3))) i4v s_i4v;   // LDS int4

__device__ __forceinline__ void async_cp16(const float* g, float* l) {
    __builtin_amdgcn_global_load_async_to_lds_b128(
        (g_i4v*)(unsigned long long)g,
        (s_i4v*)(unsigned)(unsigned long long)l,
        0, 0);
}
#else
__device__ __forceinline__ void async_cp16(const float* g, float* l) {
    asm volatile("global_load_async_to_lds_b128 %0, %1, off"
                 :: "v"((unsigned)(unsigned long long)l), "v"(g)
                 : "memory");
}
#endif

__device__ __forceinline__ void wait_async0() {
#if __has_builtin(__builtin_amdgcn_s_wait_asynccnt)
    __builtin_amdgcn_s_wait_asynccnt(0);
#else
    asm volatile("s_wait_asynccnt 0x0" ::: "memory");
#endif
}

__global__ __launch_bounds__(256, 1)
void attn_fwd_f32_wmma(const float* __restrict__ Q, const float* __restrict__ K,
                       const float* __restrict__ V, const int* __restrict__ vlen,
                       float* __restrict__ O, int Lq, int Lk)
{
    __shared__ float sK[2][KC * KSTRIDE];
    __shared__ float sV[2][KC * KSTRIDE];
    __shared__ float sP[WAVES][16 * PSTRIDE];

    const int b     = blockIdx.y;
    const int qtile = blockIdx.x;
    const int tid   = threadIdx.x;
    const int wave  = tid >> 5;
    const int lane  = tid & 31;
    const int half  = lane >> 4;   // which 16-lane half of the wave
    const int l16   = lane & 15;

    const int   vl    = vlen[b];
    const float scale = 0.08838834764831845f; // 1/sqrt(128)

    // ---- load Q tile (16 rows) register-resident in WMMA A-operand layout ----
    // A 16x4 f32 layout: lanes 0-15 hold M=lane, K={k0,k0+1}; lanes 16-31 hold M=lane-16, K={k0+2,k0+3}
    const int    qbase = qtile * MT + wave * 16;
    const size_t qoff  = ((size_t)b * Lq + (qbase + l16)) * D_DIM;

    v2f qa[32];
#pragma unroll
    for (int c = 0; c < 32; ++c)
        qa[c] = *(const v2f*)(Q + qoff + 4 * c + 2 * half);

    // ---- output accumulators: 8 d-tiles of 16x16, and softmax state ----
    v8f acc[8];
#pragma unroll
    for (int t = 0; t < 8; ++t)
#pragma unroll
        for (int r = 0; r < 8; ++r) acc[t][r] = 0.0f;

    float mrow[8], lrow[8];
#pragma unroll
    for (int r = 0; r < 8; ++r) { mrow[r] = -3.0e38f; lrow[r] = 0.0f; }

    // cooperative K/V chunk loader mapping: 8 threads per 128-float row, 16 floats each
    const int    ldr = tid >> 3;
    const int    ldc = (tid & 7) * 16;
    const float* Kb  = K + (size_t)b * Lk * D_DIM;
    const float* Vb  = V + (size_t)b * Lk * D_DIM;

    // number of key-chunks actually needed (vl is per-batch -> uniform across block)
    const int kend = min(Lk, ((vl + KC - 1) / KC) * KC);

    // ---- prefetch chunk 0 into buffer 0 ----
    {
        const float* gk = Kb + (size_t)ldr * D_DIM + ldc;
        const float* gv = Vb + (size_t)ldr * D_DIM + ldc;
        float* dk = &sK[0][ldr * KSTRIDE + ldc];
        float* dv = &sV[0][ldr * KSTRIDE + ldc];
#pragma unroll
        for (int i = 0; i < 4; ++i) {
            async_cp16(gk + 4 * i, dk + 4 * i);
            async_cp16(gv + 4 * i, dv + 4 * i);
        }
    }

    int cur = 0;
    for (int kb = 0; kb < kend; kb += KC, cur ^= 1) {
        wait_async0();       // this wave's pending copies (current chunk) have landed in LDS
        __syncthreads();     // all waves' copies landed; prior iteration's reads all done

        // ---- prefetch next chunk into the other buffer (overlaps compute below) ----
        if (kb + KC < kend) {
            const float* gk = Kb + (size_t)(kb + KC + ldr) * D_DIM + ldc;
            const float* gv = Vb + (size_t)(kb + KC + ldr) * D_DIM + ldc;
            float* dk = &sK[cur ^ 1][ldr * KSTRIDE + ldc];
            float* dv = &sV[cur ^ 1][ldr * KSTRIDE + ldc];
#pragma unroll
            for (int i = 0; i < 4; ++i) {
                async_cp16(gk + 4 * i, dk + 4 * i);
                async_cp16(gv + 4 * i, dv + 4 * i);
            }
        }

        const float* kbuf = &sK[cur][0];
        const float* vbuf = &sV[cur][0];

        // ---- S = Q * Kchunk^T : two 16x16 tiles (keys 0-15, 16-31), pipelined B loads ----
        v8f s0, s1;
#pragma unroll
        for (int r = 0; r < 8; ++r) { s0[r] = 0.0f; s1[r] = 0.0f; }

        v2f b0 = *(const v2f*)(kbuf + l16 * KSTRIDE + 2 * half);
        v2f b1 = *(const v2f*)(kbuf + (16 + l16) * KSTRIDE + 2 * half);
#pragma unroll
        for (int c = 0; c < 32; ++c) {
            v2f nb0 = b0, nb1 = b1;
            if (c < 31) {
                nb0 = *(const v2f*)(kbuf + l16 * KSTRIDE + 4 * (c + 1) + 2 * half);
                nb1 = *(const v2f*)(kbuf + (16 + l16) * KSTRIDE + 4 * (c + 1) + 2 * half);
            }
            s0 = __builtin_amdgcn_wmma_f32_16x16x4_f32(false, qa[c], false, b0,
                                                       (short)0, s0, false, false);
            s1 = __builtin_amdgcn_wmma_f32_16x16x4_f32(false, qa[c], false, b1,
                                                       (short)0, s1, false, false);
            b0 = nb0; b1 = nb1;
        }

        // ---- scale + key mask (C layout: element (M=r+8*half, N=l16)) ----
        const bool ok0 = (kb + l16) < vl;
        const bool ok1 = (kb + 16 + l16) < vl;
#pragma unroll
        for (int r = 0; r < 8; ++r) {
            s0[r] = ok0 ? s0[r] * scale : NEGVAL;
            s1[r] = ok1 ? s1[r] * scale : NEGVAL;
        }

        // ---- online softmax: row max / rescale / exp / row sum ----
        float alpha[8];
#pragma unroll
        for (int r = 0; r < 8; ++r) {
            float mx = fmaxf(s0[r], s1[r]);
#pragma unroll
            for (int off = 8; off > 0; off >>= 1)
                mx = fmaxf(mx, __shfl_xor(mx, off, 32));   // stays within 16-lane half
            float mn = fmaxf(mrow[r], mx);
            alpha[r] = __expf(mrow[r] - mn);
            mrow[r]  = mn;
        }
#pragma unroll
        for (int r = 0; r < 8; ++r) {
            s0[r] = __expf(s0[r] - mrow[r]);
            s1[r] = __expf(s1[r] - mrow[r]);
            float rs = s0[r] + s1[r];
#pragma unroll
            for (int off = 8; off > 0; off >>= 1)
                rs += __shfl_xor(rs, off, 32);
            lrow[r] = lrow[r] * alpha[r] + rs;
        }
#pragma unroll
        for (int t = 0; t < 8; ++t)
#pragma unroll
            for (int r = 0; r < 8; ++r) acc[t][r] *= alpha[r];

        // ---- P: C-layout -> A-layout via per-wave LDS (same-wave LDS is in-order) ----
        float* pw = &sP[wave][0];
#pragma unroll
        for (int r = 0; r < 8; ++r) {
            pw[(r + 8 * half) * PSTRIDE + l16]      = s0[r];
            pw[(r + 8 * half) * PSTRIDE + 16 + l16] = s1[r];
        }

        // ---- O += P * Vchunk (pipelined A/B loads) ----
#pragma unroll
        for (int ks = 0; ks < 8; ++ks) {
            const int k0 = 4 * ks;
            v2f ap = *(const v2f*)(pw + l16 * PSTRIDE + k0 + 2 * half);
            v2f bv;
            bv.x = vbuf[(k0 + 2 * half) * KSTRIDE + l16];
            bv.y = vbuf[(k0 + 2 * half + 1) * KSTRIDE + l16];
#pragma unroll
            for (int t = 0; t < 8; ++t) {
                v2f nbv = bv;
                if (t < 7) {
                    nbv.x = vbuf[(k0 + 2 * half) * KSTRIDE + (t + 1) * 16 + l16];
                    nbv.y = vbuf[(k0 + 2 * half + 1) * KSTRIDE + (t + 1) * 16 + l16];
                }
                acc[t] = __builtin_amdgcn_wmma_f32_16x16x4_f32(false, ap, false, bv,
                                                               (short)0, acc[t], false, false);
                bv = nbv;
            }
        }
    }

    // ---- epilogue: normalize by row sums, store ----
    float inv[8];
#pragma unroll
    for (int r = 0; r < 8; ++r) inv[r] = 1.0f / lrow[r];

    float* Ob = O + ((size_t)b * Lq + qbase) * D_DIM;
#pragma unroll
    for (int t = 0; t < 8; ++t)
#pragma unroll
        for (int r = 0; r < 8; ++r)
            Ob[(size_t)(r + 8 * half) * D_DIM + t * 16 + l16] = acc[t][r] * inv[r];
}

extern "C" void kernel_launch(void* const* d_in, const int* in_sizes, int n_in,
                              void* d_out, int out_size, void* d_ws, size_t ws_size,
                              hipStream_t stream) {
    const float* Q  = (const float*)d_in[0];
    const float* K  = (const float*)d_in[1];
    const float* V  = (const float*)d_in[2];
    const int*   vl = (const int*)d_in[3];
    float*       O  = (float*)d_out;

    const int B  = in_sizes[3];
    const int Lq = in_sizes[0] / (B * D_DIM);
    const int Lk = in_sizes[1] / (B * D_DIM);

    dim3 grid(Lq / MT, B);
    attn_fwd_f32_wmma<<<grid, 256, 0, stream>>>(Q, K, V, vl, O, Lq, Lk);
}